// Averager_45509473468765
// MI455X (gfx1250) — compile-verified
//
#include <hip/hip_runtime.h>

// MI455X (gfx1250): bandwidth-bound permute/pool/broadcast (~200MB traffic,
// ~8.6us at 23.3TB/s). Level-1 channel pooling expressed as exact-f32
// V_WMMA_F32_16X16X4_F32 chain (16 wmma per 256x64 tile).

typedef __attribute__((ext_vector_type(2))) float v2f;
typedef __attribute__((ext_vector_type(8))) float v8f;

#define B_  32
#define N_  4096

// ---------------- Level 0: out[:,:,0,:] = x[:,:,0,:] (coalesced copy) -----
__global__ __launch_bounds__(256) void lvl0_copy(const float4* __restrict__ in4,
                                                 float4* __restrict__ out4) {
    const int total = B_ * N_ * 16;              // float4s in the i=0 slice
    int gid = blockIdx.x * blockDim.x + threadIdx.x;
    int stride = gridDim.x * blockDim.x;
    for (int F = gid; F < total; F += stride) {
        int token = F >> 4;
        int g     = F & 15;
        int a4 = token * 48 + g;                 // (token*3+0)*64/4 + g
        out4[a4] = in4[a4];
    }
}

// ---------------- Level 1: per (b, T) tile, V = P*X*Q / 64 ----------------
// T = 4*(tz&3)+(ty&3); tile = x[b, T*256:(T+1)*256, 1, :]  (256 tokens x 64 ch)
// W[r, c'] = sum over 16 tokens of group r (r = 4*z1 + x1)      [VALU]
// V[r, q]  = sum_{cc} W[r, 4q+cc]                                [WMMA f32 16x16x4]
// broadcast each V[r,q] to 64 output positions.
__global__ __launch_bounds__(256) void lvl1_kernel(const float* __restrict__ in,
                                                   float* __restrict__ out) {
    __shared__ __align__(16) float W[16][64];
    __shared__ __align__(16) float V[16][16];

    const int blk = blockIdx.x;
    const int b   = blk >> 4;
    const int T   = blk & 15;
    const int t   = threadIdx.x;

    // ---- phase 1: token pooling (P*X), thread (r, c16) owns W[r][4c16..+3]
    const int r   = t >> 4;
    const int c16 = t & 15;
    const int z1  = r >> 2;
    const int x1  = r & 3;

    const float4* in4 = (const float4*)in;
    const int tokbase = b * N_ + T * 256;
    float4 w4 = make_float4(0.f, 0.f, 0.f, 0.f);
#pragma unroll
    for (int k = 0; k < 16; ++k) {
        int a_ = k >> 2;
        int b2 = k & 3;
        int m  = (4 * z1 + a_) * 16 + 4 * b2 + x1;      // local token
        float4 v = in4[(tokbase + m) * 48 + 16 + c16];  // i=1 slice
        w4.x += v.x; w4.y += v.y; w4.z += v.z; w4.w += v.w;
    }
    W[r][4 * c16 + 0] = w4.x;
    W[r][4 * c16 + 1] = w4.y;
    W[r][4 * c16 + 2] = w4.z;
    W[r][4 * c16 + 3] = w4.w;
    __syncthreads();

    // ---- phase 2: V = W * Q via 16 chained V_WMMA_F32_16X16X4_F32 (wave 0)
    if (t < 32) {
        const int L  = t;
        const int M  = L & 15;            // A row / B col / D col
        const int Kb = (L >> 4) * 2;      // A K-pair base for this lane half
        v8f acc = {0.f, 0.f, 0.f, 0.f, 0.f, 0.f, 0.f, 0.f};
#pragma unroll
        for (int k4 = 0; k4 < 16; ++k4) {
            v2f a;
            a.x = W[M][4 * k4 + Kb];
            a.y = W[M][4 * k4 + Kb + 1];
            float bv = (M == k4) ? 1.0f : 0.0f;   // Q chunk: col k4 all-ones
            v2f bb;
            bb.x = bv;
            bb.y = bv;
            acc = __builtin_amdgcn_wmma_f32_16x16x4_f32(
                false, a, false, bb, (short)0, acc, false, false);
        }
        const int Mo = (L >> 4) * 8;      // D layout: vgpr v -> row v + half*8
#pragma unroll
        for (int v = 0; v < 8; ++v)
            V[Mo + v][M] = acc[v] * (1.0f / 64.0f);
    }
    __syncthreads();

    // ---- phase 3: broadcast writes. value (r=4z1+x1, q=4txl+y1) -> 64 slots
    const int tzl = T >> 2;
    const int tyl = T & 3;
    const int tzh = t >> 6;
    const int tyh = (t >> 4) & 3;
    const int tx  = t & 15;
    const int txl = tx & 3;
    const int n   = (tzh * 4 + tzl) * 256 + (tyh * 4 + tyl) * 16 + tx;
    float4* out4 = (float4*)out;
    const int ob = (b * N_ + n) * 48 + 16;          // i=1 slice, float4 units
#pragma unroll
    for (int z1o = 0; z1o < 4; ++z1o) {
#pragma unroll
        for (int y1o = 0; y1o < 4; ++y1o) {
            float4 val;
            val.x = V[4 * z1o + 0][txl * 4 + y1o];
            val.y = V[4 * z1o + 1][txl * 4 + y1o];
            val.z = V[4 * z1o + 2][txl * 4 + y1o];
            val.w = V[4 * z1o + 3][txl * 4 + y1o];
            out4[ob + z1o * 4 + y1o] = val;         // c = 16z1+4y1+x1
        }
    }
}

// ---------------- Level 2: 64 values per batch, broadcast to all tokens ---
// out[b,n,2, 16z1+4y1+x1] = (1/4096) * sum over {n': n'>>10==z1, (n'>>4)&3==x1}
//                                      of sum_{c' in [16y1,16y1+16)} x[b,n',2,c']
__global__ __launch_bounds__(1024) void lvl2_kernel(const float* __restrict__ in,
                                                    float* __restrict__ out) {
    __shared__ float Vsum[64];
    const int b = blockIdx.x;
    const int t = threadIdx.x;
    if (t < 64) Vsum[t] = 0.f;
    __syncthreads();

    const float4* in4 = (const float4*)in;
    const int g  = t & 15;          // channel float4 -> y1 = g>>2 (fixed)
    const int y1 = g >> 2;
    const int x1 = t >> 8;          // (n>>4)&3 for all this thread's tokens
    const int nlo = t >> 4;         // low 6 bits of token index

#pragma unroll
    for (int z1 = 0; z1 < 4; ++z1) {
        float4 a4 = make_float4(0.f, 0.f, 0.f, 0.f);
#pragma unroll
        for (int i2 = 0; i2 < 16; ++i2) {
            int n = (z1 * 16 + i2) * 64 + nlo;              // coalesced F = it*1024+t
            float4 v = in4[(b * N_ + n) * 48 + 32 + g];     // i=2 slice
            a4.x += v.x; a4.y += v.y; a4.z += v.z; a4.w += v.w;
        }
        atomicAdd(&Vsum[z1 * 16 + y1 * 4 + x1], a4.x + a4.y + a4.z + a4.w);
    }
    __syncthreads();

    float4 val;
    val.x = Vsum[4 * g + 0] * (1.0f / 4096.0f);
    val.y = Vsum[4 * g + 1] * (1.0f / 4096.0f);
    val.z = Vsum[4 * g + 2] * (1.0f / 4096.0f);
    val.w = Vsum[4 * g + 3] * (1.0f / 4096.0f);

    float4* out4 = (float4*)out;
#pragma unroll 8
    for (int it = 0; it < 64; ++it) {
        int n = it * 64 + nlo;
        out4[(b * N_ + n) * 48 + 32 + g] = val;
    }
}

extern "C" void kernel_launch(void* const* d_in, const int* in_sizes, int n_in,
                              void* d_out, int out_size, void* d_ws, size_t ws_size,
                              hipStream_t stream) {
    const float* x = (const float*)d_in[0];
    float* out = (float*)d_out;
    (void)in_sizes; (void)n_in; (void)out_size; (void)d_ws; (void)ws_size;

    lvl0_copy<<<2048, 256, 0, stream>>>((const float4*)x, (float4*)out);
    lvl1_kernel<<<512, 256, 0, stream>>>(x, out);
    lvl2_kernel<<<32, 1024, 0, stream>>>(x, out);
}